// KMISPooling_58368605552746
// MI455X (gfx1250) — compile-verified
//
#include <hip/hip_runtime.h>
#include <stdint.h>

typedef unsigned long long u64;
typedef unsigned int u32;

#define TPB 256
#define BTILE 2048        // bitonic shared-memory tile (u64 elements)
#define ROUNDS 64

// scalar slots in ws
#define SC_DONE 0
#define SC_ZERO 1
#define SC_M    2
#define SC_U    3
#define SC_EP   4
#define SC_CNT0 8        // 64 round counters
#define SC_TOTAL 128

// ---------------------------------------------------------------------------
// init
// ---------------------------------------------------------------------------
__global__ void k_init_scalars(u32* sc, int n) {
    int i = blockIdx.x * blockDim.x + threadIdx.x;
    if (i < n) sc[i] = 0u;
}

__global__ void k_build_node_keys(const float* __restrict__ score, u64* __restrict__ buf,
                                  int N, int Npad) {
    int i = blockIdx.x * blockDim.x + threadIdx.x;
    if (i >= Npad) return;
    if (i < N) {
        u32 u = __float_as_uint(score[i]);
        u32 ks = (u & 0x80000000u) ? ~u : (u ^ 0x80000000u);   // monotone float->uint
        buf[i] = ((u64)ks << 32) | (u32)i;                      // stable tie-break by index
    } else {
        buf[i] = ~0ull;
    }
}

__global__ void k_rank_scatter(const u64* __restrict__ buf, int* __restrict__ rank, int N) {
    int t = blockIdx.x * blockDim.x + threadIdx.x;
    if (t >= N) return;
    u32 idx = (u32)(buf[t] & 0xFFFFFFFFu);
    rank[idx] = t;
}

__global__ void k_mis_init(const int* __restrict__ rank, int* A, int* B,
                           int* maskA, int* maskB, int* mis, int N) {
    int v = blockIdx.x * blockDim.x + threadIdx.x;
    if (v >= N) return;
    int r = rank[v];
    A[v] = r; B[v] = r;
    maskA[v] = 0; maskB[v] = 0; mis[v] = 0;
}

// ---------------------------------------------------------------------------
// bitonic sort (u64 ascending) : shared full-sort tile + global merges + shared merge
// ---------------------------------------------------------------------------
__global__ __launch_bounds__(1024) void k_bsort_shared(u64* __restrict__ a) {
    __shared__ u64 s[BTILE];
    int base = blockIdx.x * BTILE;
    int t = threadIdx.x;
    s[t] = a[base + t];
    s[t + 1024] = a[base + t + 1024];
    __syncthreads();
    for (int k = 2; k <= BTILE; k <<= 1) {
        for (int j = k >> 1; j > 0; j >>= 1) {
            int i = ((t & ~(j - 1)) << 1) | (t & (j - 1));
            int p = i | j;
            bool up = ((((u32)(base + i)) & (u32)k) == 0u);
            u64 x = s[i], y = s[p];
            if ((x > y) == up) { s[i] = y; s[p] = x; }
            __syncthreads();
        }
    }
    a[base + t] = s[t];
    a[base + t + 1024] = s[t + 1024];
}

__global__ __launch_bounds__(1024) void k_bmerge_shared(u64* __restrict__ a, int k) {
    __shared__ u64 s[BTILE];
    int base = blockIdx.x * BTILE;
    int t = threadIdx.x;
    s[t] = a[base + t];
    s[t + 1024] = a[base + t + 1024];
    __syncthreads();
    for (int j = 1024; j > 0; j >>= 1) {
        int i = ((t & ~(j - 1)) << 1) | (t & (j - 1));
        int p = i | j;
        bool up = ((((u32)(base + i)) & (u32)k) == 0u);
        u64 x = s[i], y = s[p];
        if ((x > y) == up) { s[i] = y; s[p] = x; }
        __syncthreads();
    }
    a[base + t] = s[t];
    a[base + t + 1024] = s[t + 1024];
}

__global__ void k_bmerge_global(u64* __restrict__ a, int j, int k, int half) {
    int t = blockIdx.x * blockDim.x + threadIdx.x;
    if (t >= half) return;
    int i = ((t & ~(j - 1)) << 1) | (t & (j - 1));
    int p = i | j;
    bool up = ((((u32)i) & (u32)k) == 0u);
    u64 x = a[i], y = a[p];
    if ((x > y) == up) { a[i] = y; a[p] = x; }
}

// ---------------------------------------------------------------------------
// MIS rounds (K=1): scatter-min, mis/mask update, scatter-or, reset+count
// ---------------------------------------------------------------------------
__global__ void k_edge_min(const int* __restrict__ ei, int E,
                           const int* __restrict__ A, int* __restrict__ B,
                           const u32* __restrict__ flag) {
    if (*flag) return;
    int e = blockIdx.x * blockDim.x + threadIdx.x;
    if (e >= E) return;
    if (e + 8192 < E) {
        __builtin_prefetch(ei + e + 8192, 0, 0);           // -> global_prefetch_b8
        __builtin_prefetch(ei + E + e + 8192, 0, 0);
    }
    int r = ei[e];
    int c = ei[E + e];
    atomicMin(&B[c], A[r]);
}

__global__ void k_mis_mask(const int* __restrict__ rank, const int* __restrict__ B,
                           int* maskA, int* maskB, int* mis, int N,
                           const u32* __restrict__ flag) {
    if (*flag) return;
    int v = blockIdx.x * blockDim.x + threadIdx.x;
    if (v >= N) return;
    int m = maskA[v];
    int mi = mis[v];
    if (!m && rank[v] == B[v]) mi = 1;
    mis[v] = mi;
    int nm = m | mi;
    maskA[v] = nm;
    maskB[v] = nm;
}

__global__ void k_edge_mask(const int* __restrict__ ei, int E,
                            const int* __restrict__ maskA, int* __restrict__ maskB,
                            const u32* __restrict__ flag) {
    if (*flag) return;
    int e = blockIdx.x * blockDim.x + threadIdx.x;
    if (e >= E) return;
    if (e + 8192 < E) {
        __builtin_prefetch(ei + e + 8192, 0, 0);
        __builtin_prefetch(ei + E + e + 8192, 0, 0);
    }
    int r = ei[e];
    int c = ei[E + e];
    if (maskA[r]) maskB[c] = 1;   // idempotent store of 1 (order-independent)
}

__global__ void k_reset(const int* __restrict__ rank, int* A, int* B,
                        int* maskA, const int* __restrict__ maskB, int N,
                        u32* cnt, const u32* __restrict__ flag) {
    if (*flag) return;
    int v = blockIdx.x * blockDim.x + threadIdx.x;
    int m = 1;
    if (v < N) {
        m = maskB[v];
        maskA[v] = m;
        int val = m ? N : rank[v];
        A[v] = val; B[v] = val;
    }
    unsigned long long bal = __ballot(v < N && !m);         // wave32 ballot
    if ((threadIdx.x & (warpSize - 1)) == 0 && bal)
        atomicAdd(cnt, (u32)__popcll(bal));
}

__global__ void k_check(const u32* cnt, u32* flag) {
    if (*cnt == 0u) *flag = 1u;
}

// ---------------------------------------------------------------------------
// clustering
// ---------------------------------------------------------------------------
__global__ void k_cluster_init(const int* __restrict__ rank, const int* __restrict__ mis,
                               int* A, int* B, int N) {
    int v = blockIdx.x * blockDim.x + threadIdx.x;
    if (v >= N) return;
    int val = mis[v] ? rank[v] : N;
    A[v] = val; B[v] = val;
}

__global__ void k_compact(const int* __restrict__ mis, const u32* __restrict__ pos,
                          const int* __restrict__ rank,
                          int* misNodes, int* posOfRank, int N) {
    int v = blockIdx.x * blockDim.x + threadIdx.x;
    if (v >= N) return;
    if (mis[v]) {
        u32 p = pos[v] - 1u;          // inclusive scan -> compact position
        misNodes[p] = v;
        posOfRank[rank[v]] = (int)p;
    }
}

__global__ void k_assign_cluster(const int* __restrict__ B, const int* __restrict__ posOfRank,
                                 int* __restrict__ cluster, int N) {
    int v = blockIdx.x * blockDim.x + threadIdx.x;
    if (v >= N) return;
    cluster[v] = posOfRank[B[v]];
}

// ---------------------------------------------------------------------------
// edge coarsening
// ---------------------------------------------------------------------------
__global__ void k_build_edge_keys(const int* __restrict__ ei, const int* __restrict__ cluster,
                                  u64* __restrict__ buf, int E, int Epad) {
    int e = blockIdx.x * blockDim.x + threadIdx.x;
    if (e >= Epad) return;
    if (e < E) {
        u64 cr = (u64)(u32)cluster[ei[e]];
        u64 cc = (u64)(u32)cluster[ei[E + e]];
        buf[e] = (cr << 39) | (cc << 22) | (u64)(u32)e;   // 17+17+22 bits, stable by edge id
    } else {
        buf[e] = ~0ull;
    }
}

__global__ void k_zero_seg(u32* keep, u64* sums, int E) {
    int i = blockIdx.x * blockDim.x + threadIdx.x;
    if (i >= E) return;
    keep[i] = 0u;
    sums[i] = 0ull;
}

__global__ void k_seg_accum(const u64* __restrict__ keys, const float* __restrict__ attr,
                            const u32* __restrict__ uidx, u32* __restrict__ keep,
                            u64* __restrict__ sums, int E) {
    int i = blockIdx.x * blockDim.x + threadIdx.x;
    if (i >= E) return;
    u64 ki = keys[i];
    u32 u = uidx[i] - 1u;
    int e = (int)(ki & 0x3FFFFFu);
    // deterministic fixed-point accumulation (2^-32 LSB)
    unsigned long long q = (unsigned long long)((double)attr[e] * 4294967296.0 + 0.5);
    atomicAdd(&sums[u], q);
    bool head = (i == 0) || ((ki >> 22) != (keys[i - 1] >> 22));
    if (head) {
        u32 cr = (u32)(ki >> 39);
        u32 cc = (u32)((ki >> 22) & 0x1FFFFu);
        keep[u] = (cr != cc) ? 1u : 0u;
    }
}

__global__ void k_emit_edges(const u64* __restrict__ keys, const u32* __restrict__ uidx,
                             const u32* __restrict__ kidx, const u64* __restrict__ sums,
                             float* __restrict__ out, const u32* __restrict__ sc,
                             int E, int D) {
    int i = blockIdx.x * blockDim.x + threadIdx.x;
    if (i >= E) return;
    u64 ki = keys[i];
    bool head = (i == 0) || ((ki >> 22) != (keys[i - 1] >> 22));
    if (!head) return;
    u32 cr = (u32)(ki >> 39);
    u32 cc = (u32)((ki >> 22) & 0x1FFFFu);
    if (cr == cc) return;
    u32 u = uidx[i] - 1u;
    u32 j = kidx[u] - 1u;
    u32 M = sc[SC_M], Ep = sc[SC_EP];
    size_t b1 = (size_t)M * (size_t)D;
    out[b1 + j] = (float)cr;
    out[b1 + (size_t)Ep + j] = (float)cc;
    out[b1 + 2 * (size_t)Ep + j] = (float)((double)sums[u] * (1.0 / 4294967296.0));
}

__global__ void k_emit_nodes(const int* __restrict__ cluster, const int* __restrict__ mis,
                             const float* __restrict__ score, float* __restrict__ out,
                             const u32* __restrict__ sc, int N, int D) {
    int v = blockIdx.x * blockDim.x + threadIdx.x;
    if (v >= N) return;
    u32 M = sc[SC_M], Ep = sc[SC_EP];
    size_t b = (size_t)M * (size_t)D + 3 * (size_t)Ep;
    out[b + v] = (float)cluster[v];
    out[b + (size_t)N + v] = mis[v] ? 1.0f : 0.0f;
    out[b + 2 * (size_t)N + v] = score[v];
}

// ---------------------------------------------------------------------------
// x_out row gather through the CDNA5 data movers:
//   TDM tensor_load_to_lds (D# per ISA 08 §8) -> s_wait_tensorcnt
//   (fallback: global_load_async_to_lds_b128 -> s_wait_asynccnt)
//   then global_store_async_from_lds_b128 -> s_wait_asynccnt
// ---------------------------------------------------------------------------
#if defined(__gfx1250__) && __has_builtin(__builtin_amdgcn_tensor_load_to_lds)
#define USE_TDM_GATHER 1
typedef unsigned v4u __attribute__((ext_vector_type(4)));
typedef int      v8i __attribute__((ext_vector_type(8)));
typedef int      v4i __attribute__((ext_vector_type(4)));
#endif

__global__ void k_gather_x(const float* __restrict__ x, const int* __restrict__ misNodes,
                           float* __restrict__ out, const u32* __restrict__ sc, int D) {
    int p = blockIdx.x;
    if (p >= (int)sc[SC_M]) return;
    int node = misNodes[p];
    const float* src = x + (size_t)node * (size_t)D;
    float* dst = out + (size_t)p * (size_t)D;
#if defined(__gfx1250__)
    if (D == 128) {
        __shared__ __align__(16) float lds[128];
        unsigned l = threadIdx.x;                              // one wave32, 16B per lane
        unsigned lbase = (unsigned)(size_t)(void*)lds;         // low 32 bits = LDS offset
        unsigned long long ga = (unsigned long long)(const void*)src;
#if defined(USE_TDM_GATHER)
        // D# group0: count=1 | lds_addr | global_addr[56:0] | type=2
        v4u g0 = { 1u,
                   lbase,
                   (u32)(ga & 0xFFFFFFFFull),
                   (u32)((ga >> 32) & 0x1FFFFFFull) | (2u << 30) };
        // D# group1: data_size=4B; tensor_dim0=128; tensor_dim1=1; tile_dim0=128 (1-D tile);
        //            tile_dim1/2 unused; tensor_dim0_stride=128
        v8i g1 = { (int)(2u << 16),          // workgroup_mask=0, data_size=2 (4B)
                   (int)(128u << 16),        // tensor_dim0[15:0] @ bits[63:48]
                   (int)(1u << 16),          // tensor_dim0[31:16]=0, tensor_dim1[15:0]=1
                   (int)(128u << 16),        // tensor_dim1[31:16]=0, tile_dim0=128
                   0,                        // tile_dim1=0 (unused), tile_dim2=0
                   128,                      // tensor_dim0_stride[31:0]
                   0, 0 };
        v4i gz = { 0, 0, 0, 0 };             // groups 2/3: unused (<=2-D tensor)
#if __clang_major__ >= 23
        v8i gz8 = { 0, 0, 0, 0, 0, 0, 0, 0 };
        __builtin_amdgcn_tensor_load_to_lds(g0, g1, gz, gz, gz8, 0);
#else
        __builtin_amdgcn_tensor_load_to_lds(g0, g1, gz, gz, 0);
#endif
#if __has_builtin(__builtin_amdgcn_s_wait_tensorcnt)
        __builtin_amdgcn_s_wait_tensorcnt(0);
#else
        asm volatile("s_wait_tensorcnt 0x0" ::: "memory");
#endif
#else   // no TDM builtin: per-lane async load to LDS
        unsigned laddr = lbase + l * 16u;
        unsigned long long gal = ga + (unsigned long long)l * 16ull;
        asm volatile("global_load_async_to_lds_b128 %0, %1, off"
                     :: "v"(laddr), "v"(gal) : "memory");
        asm volatile("s_wait_asynccnt 0x0" ::: "memory");
#endif
        // stream row from LDS to packed output via async store
        unsigned lsrc = lbase + l * 16u;
        unsigned long long gd = (unsigned long long)(void*)dst + (unsigned long long)l * 16ull;
        asm volatile("global_store_async_from_lds_b128 %0, %1, off"
                     :: "v"(gd), "v"(lsrc) : "memory");
        asm volatile("s_wait_asynccnt 0x0" ::: "memory");
        return;
    }
#endif
    for (int d = threadIdx.x; d < D; d += blockDim.x) dst[d] = src[d];
}

// ---------------------------------------------------------------------------
// generic 3-phase inclusive scan (mode 0: flags from in[], mode 1: segment heads from keys>>22)
// ---------------------------------------------------------------------------
__global__ void k_scan1(const u64* __restrict__ keys, const u32* __restrict__ in,
                        u32* __restrict__ out, int L, u32* __restrict__ bs, int mode) {
    __shared__ u32 sm[256];
    int t = threadIdx.x;
    int base = blockIdx.x * 1024 + t * 4;
    u32 v[4]; u32 s = 0;
#pragma unroll
    for (int q = 0; q < 4; ++q) {
        int i = base + q; u32 f = 0;
        if (i < L) {
            if (mode == 0) f = in[i] ? 1u : 0u;
            else {
                u64 ki = keys[i] >> 22;
                f = (i == 0) || (ki != (keys[i - 1] >> 22));
            }
        }
        v[q] = f; s += f;
    }
    sm[t] = s; __syncthreads();
    for (int off = 1; off < 256; off <<= 1) {
        u32 xv = (t >= off) ? sm[t - off] : 0u;
        __syncthreads();
        sm[t] += xv;
        __syncthreads();
    }
    u32 incl = sm[t];
    if (t == 255) bs[blockIdx.x] = incl;
    u32 run = incl - s;
#pragma unroll
    for (int q = 0; q < 4; ++q) {
        int i = base + q;
        run += v[q];
        if (i < L) out[i] = run;   // inclusive
    }
}

__global__ void k_scan2(u32* bs, int nb, u32* total) {
    u32 run = 0;
    for (int b = 0; b < nb; ++b) { u32 x = bs[b]; bs[b] = run; run += x; }
    *total = run;
}

__global__ void k_scan3(u32* __restrict__ out, const u32* __restrict__ bs, int L) {
    int base = blockIdx.x * 1024 + threadIdx.x * 4;
    u32 add = bs[blockIdx.x];
#pragma unroll
    for (int q = 0; q < 4; ++q) {
        int i = base + q;
        if (i < L) out[i] += add;
    }
}

// ---------------------------------------------------------------------------
// host side
// ---------------------------------------------------------------------------
static void bitonic_sort(u64* a, int n, hipStream_t s) {
    int tb = n / BTILE;
    k_bsort_shared<<<tb, 1024, 0, s>>>(a);
    for (int k = BTILE * 2; k <= n; k <<= 1) {
        for (int j = k >> 1; j >= BTILE; j >>= 1) {
            int half = n >> 1;
            k_bmerge_global<<<(half + TPB - 1) / TPB, TPB, 0, s>>>(a, j, k, half);
        }
        k_bmerge_shared<<<tb, 1024, 0, s>>>(a, k);
    }
}

static void run_scan(const u64* keys, const u32* in, u32* out, int L,
                     u32* bs, u32* total, int mode, hipStream_t s) {
    int nb = (L + 1023) / 1024;
    k_scan1<<<nb, 256, 0, s>>>(keys, in, out, L, bs, mode);
    k_scan2<<<1, 1, 0, s>>>(bs, nb, total);
    k_scan3<<<nb, 256, 0, s>>>(out, bs, L);
}

extern "C" void kernel_launch(void* const* d_in, const int* in_sizes, int n_in,
                              void* d_out, int out_size, void* d_ws, size_t ws_size,
                              hipStream_t stream) {
    (void)n_in; (void)out_size;
    const float* x     = (const float*)d_in[0];
    const int*   ei    = (const int*)d_in[1];
    const float* attr  = (const float*)d_in[2];
    const float* score = (const float*)d_in[3];

    const int ND = in_sizes[0];
    const int E  = in_sizes[2];
    const int N  = in_sizes[3];
    const int D  = ND / N;

    auto npow2 = [](int v) { int p = 1; while (p < v) p <<= 1; return p; };
    int Npad = npow2(N); if (Npad < BTILE) Npad = BTILE;
    int Epad = npow2(E); if (Epad < BTILE) Epad = BTILE;
    int SPad = (Npad > Epad) ? Npad : Epad;

    // ws layout (u64 arrays first for alignment)
    char* w = (char*)d_ws;
    u64* sortbuf = (u64*)w;            w += (size_t)SPad * 8;
    u64* sums    = (u64*)w;            w += (size_t)E * 8;
    u32* uidx    = (u32*)w;            w += (size_t)E * 4;
    u32* keep    = (u32*)w;            w += (size_t)E * 4;
    u32* kidx    = (u32*)w;            w += (size_t)E * 4;
    int* rank      = (int*)w;          w += (size_t)N * 4;
    int* A         = (int*)w;          w += (size_t)N * 4;   // min_rank (read)
    int* B         = (int*)w;          w += (size_t)N * 4;   // min_rank (accumulate)
    int* maskA     = (int*)w;          w += (size_t)N * 4;
    int* maskB     = (int*)w;          w += (size_t)N * 4;
    int* mis       = (int*)w;          w += (size_t)N * 4;
    u32* pos       = (u32*)w;          w += (size_t)N * 4;
    int* misNodes  = (int*)w;          w += (size_t)N * 4;
    int* posOfRank = (int*)w;          w += (size_t)N * 4;
    int* cluster   = (int*)w;          w += (size_t)N * 4;
    u32* bsums     = (u32*)w;          w += (size_t)8192 * 4;
    u32* sc        = (u32*)w;          w += (size_t)SC_TOTAL * 4;

    size_t need = (size_t)(w - (char*)d_ws);
    if (ws_size < need) return;   // insufficient scratch: do nothing (deterministic)

    const int nbN = (N + TPB - 1) / TPB;
    const int nbE = (E + TPB - 1) / TPB;
    float* out = (float*)d_out;

    // --- init + rank = argsort(argsort(score)) via stable bitonic key sort ---
    k_init_scalars<<<1, SC_TOTAL, 0, stream>>>(sc, SC_TOTAL);
    k_build_node_keys<<<(Npad + TPB - 1) / TPB, TPB, 0, stream>>>(score, sortbuf, N, Npad);
    bitonic_sort(sortbuf, Npad, stream);
    k_rank_scatter<<<nbN, TPB, 0, stream>>>(sortbuf, rank, N);
    k_mis_init<<<nbN, TPB, 0, stream>>>(rank, A, B, maskA, maskB, mis, N);

    // --- maximal 1-independent-set: fixed 64 rounds, device-side early exit ---
    for (int r = 0; r < ROUNDS; ++r) {
        k_edge_min<<<nbE, TPB, 0, stream>>>(ei, E, A, B, sc + SC_DONE);
        k_mis_mask<<<nbN, TPB, 0, stream>>>(rank, B, maskA, maskB, mis, N, sc + SC_DONE);
        k_edge_mask<<<nbE, TPB, 0, stream>>>(ei, E, maskA, maskB, sc + SC_DONE);
        k_reset<<<nbN, TPB, 0, stream>>>(rank, A, B, maskA, maskB, N, sc + SC_CNT0 + r, sc + SC_DONE);
        k_check<<<1, 1, 0, stream>>>(sc + SC_CNT0 + r, sc + SC_DONE);
    }

    // --- cluster assignment: 1-hop scatter-min of MIS ranks ---
    k_cluster_init<<<nbN, TPB, 0, stream>>>(rank, mis, A, B, N);
    k_edge_min<<<nbE, TPB, 0, stream>>>(ei, E, A, B, sc + SC_ZERO);  // flag stays 0
    run_scan(nullptr, (const u32*)mis, pos, N, bsums, sc + SC_M, 0, stream);
    k_compact<<<nbN, TPB, 0, stream>>>(mis, pos, rank, misNodes, posOfRank, N);
    k_assign_cluster<<<nbN, TPB, 0, stream>>>(B, posOfRank, cluster, N);

    // --- coarsened edges: sort packed (cr,cc,e) keys, segment-sum, drop diagonal ---
    k_build_edge_keys<<<(Epad + TPB - 1) / TPB, TPB, 0, stream>>>(ei, cluster, sortbuf, E, Epad);
    bitonic_sort(sortbuf, Epad, stream);
    k_zero_seg<<<nbE, TPB, 0, stream>>>(keep, sums, E);
    run_scan(sortbuf, nullptr, uidx, E, bsums, sc + SC_U, 1, stream);
    k_seg_accum<<<nbE, TPB, 0, stream>>>(sortbuf, attr, uidx, keep, sums, E);
    run_scan(nullptr, keep, kidx, E, bsums, sc + SC_EP, 0, stream);

    // --- emit outputs: [x_out | edge_index_out(2,E') | edge_attr_out | cluster | mis | score] ---
    k_emit_edges<<<nbE, TPB, 0, stream>>>(sortbuf, uidx, kidx, sums, out, sc, E, D);
    k_gather_x<<<N, 32, 0, stream>>>(x, misNodes, out, sc, D);
    k_emit_nodes<<<nbN, TPB, 0, stream>>>(cluster, mis, score, out, sc, N, D);
}